// VariableClementsCDiag1_6047313953019
// MI455X (gfx1250) — compile-verified
//
#include <hip/hip_runtime.h>
#include <hip/hip_bf16.h>
#include <math.h>

// MI455X / gfx1250, wave32. Strategy:
//   out = X @ U^T  where U = diag(e^{i*omega}) * L128 * ... * L1  (512x512 complex)
// Kernel 1 builds U by sweeping the 128 MZI layers over identity columns
// (serial part, 1/4 the work of simulating the batch directly), storing
// PLANAR Ure[f][g], Uim[f][g] so GEMM B-fragments are contiguous b64 loads.
// Kernel 2 is a dense complex GEMM on the matrix cores using the native-F32
// V_WMMA_F32_16X16X4_F32 (keeps full f32 precision like the reference).

typedef float v2f __attribute__((ext_vector_type(2)));
typedef float v8f __attribute__((ext_vector_type(8)));

#define NFEAT 512
#define NPAIR 256
#define NSTEP 64
#define BATCH 2048

struct C2 { float re, im; };
__device__ __forceinline__ C2 cmul(C2 a, C2 b) { return { a.re*b.re - a.im*b.im, a.re*b.im + a.im*b.re }; }
__device__ __forceinline__ C2 cadd(C2 a, C2 b) { return { a.re + b.re, a.im + b.im }; }
__device__ __forceinline__ C2 csub(C2 a, C2 b) { return { a.re - b.re, a.im - b.im }; }
__device__ __forceinline__ C2 imul(C2 z)       { return { -z.im, z.re }; }   // i*z

// One MZI: na = ephi*t*a + i*u*b ; nb = i*ephi*u*a - t*b
__device__ __forceinline__ void mzi(C2& a, C2& b, float ang0, float ang1) {
    float s0, c0, s1, c1;
    sincosf(ang0, &s0, &c0);
    sincosf(ang1, &s1, &c1);
    C2 ephi = { c0, s0 };
    C2 tt   = { 0.5f * (c1 - 1.0f), 0.5f * s1 };
    C2 uu   = { 0.5f * (c1 + 1.0f), 0.5f * s1 };
    C2 et = cmul(ephi, tt);
    C2 eu = cmul(ephi, uu);
    C2 na = cadd(cmul(et, a), cmul(imul(uu), b));
    C2 nb = csub(imul(cmul(eu, a)), cmul(tt, b));
    a = na;
    b = nb;
}

// ---------------------------------------------------------------------------
// Kernel 1: build U, one workgroup per column g. Thread t owns features
// f=2t (ca) and f=2t+1 (cb). Layer A is register-local; layer B exchanges
// one complex value per thread through LDS. Stores planar Ure/Uim [f][g].
// ---------------------------------------------------------------------------
__global__ __launch_bounds__(NPAIR) void build_u_kernel(
    const float* __restrict__ A0, const float* __restrict__ A1,
    const float* __restrict__ B0, const float* __restrict__ B1,
    const float* __restrict__ omega,
    float* __restrict__ Ure,   // [512 f][512 g]
    float* __restrict__ Uim)   // [512 f][512 g]
{
    const int t = threadIdx.x;   // pair id 0..255
    const int g = blockIdx.x;    // column of U

    __shared__ float2 xs[NPAIR];

    // identity column of U
    C2 ca = { (2 * t     == g) ? 1.0f : 0.0f, 0.0f };
    C2 cb = { (2 * t + 1 == g) ? 1.0f : 0.0f, 0.0f };

    for (int j = 0; j < NSTEP; ++j) {
        // layer A: pairs (2t, 2t+1) -- fully thread-local
        mzi(ca, cb, A0[j * NPAIR + t], A1[j * NPAIR + t]);

        // layer B: pairs (2t+1, 2t+2) for t = 0..254
        xs[t] = make_float2(ca.re, ca.im);
        __syncthreads();
        if (t < NPAIR - 1) {
            float2 nb = xs[t + 1];
            C2 b = { nb.x, nb.y };
            C2 a = cb;
            mzi(a, b, B0[j * (NPAIR - 1) + t], B1[j * (NPAIR - 1) + t]);
            cb = a;                              // new x[2t+1]
            xs[t + 1] = make_float2(b.re, b.im); // new x[2t+2] -> thread t+1's ca
        }
        __syncthreads();
        float2 r = xs[t];
        ca = { r.x, r.y };
    }

    // final per-feature phase e^{i*omega_f}
    {
        float s, c;
        sincosf(omega[2 * t], &s, &c);
        ca = cmul(ca, (C2){ c, s });
        sincosf(omega[2 * t + 1], &s, &c);
        cb = cmul(cb, (C2){ c, s });
    }

    // planar store: U[f][g]
    Ure[(size_t)(2 * t)     * NFEAT + g] = ca.re;
    Uim[(size_t)(2 * t)     * NFEAT + g] = ca.im;
    Ure[(size_t)(2 * t + 1) * NFEAT + g] = cb.re;
    Uim[(size_t)(2 * t + 1) * NFEAT + g] = cb.im;
}

// ---------------------------------------------------------------------------
// Kernel 2: out[b][f] = sum_g X[b][g] * U[f][g]   (complex, f32)
// Each wave owns a 32x32 output block = 2x2 tiles of 16x16, accumulated with
// V_WMMA_F32_16X16X4_F32. 4 real chains per tile: P=Xr*Ur Q=Xi*Ui R=Xr*Ui S=Xi*Ur
// out_re = P - Q ; out_im = R + S.
// ---------------------------------------------------------------------------
__global__ __launch_bounds__(256) void gemm_kernel(
    const float* __restrict__ X,      // (2048, 512, 2) complex interleaved
    const float* __restrict__ Ure,    // (512 f, 512 g) planar
    const float* __restrict__ Uim,    // (512 f, 512 g) planar
    float* __restrict__ OUT)          // (2048, 512, 2)
{
    const int lane  = threadIdx.x & 31;
    const int wave  = threadIdx.x >> 5;
    const int strip = blockIdx.x * 8 + wave;      // 1024 strips total
    const int sm = strip & 63;                    // 64 strips over batch
    const int sn = strip >> 6;                    // 16 strips over features
    const int b0 = sm * 32;
    const int f0 = sn * 32;

    const int ml = lane & 15;                     // M row for A, N col for B/D
    const int kh = (lane >> 4) << 1;              // lane's K pair base: 0 or 2

    v8f P[2][2] = {{{0.f},{0.f}},{{0.f},{0.f}}};
    v8f Q[2][2] = {{{0.f},{0.f}},{{0.f},{0.f}}};
    v8f R[2][2] = {{{0.f},{0.f}},{{0.f},{0.f}}};
    v8f S[2][2] = {{{0.f},{0.f}},{{0.f},{0.f}}};

    // per-lane base addresses
    const float* xrow[2];
    const float* ur[2];
    const float* ui[2];
#pragma unroll
    for (int ti = 0; ti < 2; ++ti)
        xrow[ti] = X + ((size_t)(b0 + ti * 16 + ml) * NFEAT + kh) * 2;
#pragma unroll
    for (int tj = 0; tj < 2; ++tj) {
        const size_t frow = (size_t)(f0 + tj * 16 + ml) * NFEAT + kh;
        ur[tj] = Ure + frow;
        ui[tj] = Uim + frow;
    }

    for (int g0 = 0; g0 < NFEAT; g0 += 4) {
        // A fragments (16x4 f32): lane(l<16): M=l, K=0,1 ; lane(l>=16): K=2,3
        v2f Are[2], Aim[2];
#pragma unroll
        for (int ti = 0; ti < 2; ++ti) {
            const float4 xv = *(const float4*)(xrow[ti] + g0 * 2);
            Are[ti] = (v2f){ xv.x, xv.z };   // re at g = gk, gk+1
            Aim[ti] = (v2f){ xv.y, xv.w };   // im at g = gk, gk+1
        }

        // B fragments (4x16 f32): contiguous b64 from planar U[f][g]
        v2f Bre[2], Bim[2];
#pragma unroll
        for (int tj = 0; tj < 2; ++tj) {
            Bre[tj] = *(const v2f*)(ur[tj] + g0);
            Bim[tj] = *(const v2f*)(ui[tj] + g0);
        }

#pragma unroll
        for (int ti = 0; ti < 2; ++ti)
#pragma unroll
            for (int tj = 0; tj < 2; ++tj) {
                P[ti][tj] = __builtin_amdgcn_wmma_f32_16x16x4_f32(false, Are[ti], false, Bre[tj], (short)0, P[ti][tj], false, false);
                Q[ti][tj] = __builtin_amdgcn_wmma_f32_16x16x4_f32(false, Aim[ti], false, Bim[tj], (short)0, Q[ti][tj], false, false);
                R[ti][tj] = __builtin_amdgcn_wmma_f32_16x16x4_f32(false, Are[ti], false, Bim[tj], (short)0, R[ti][tj], false, false);
                S[ti][tj] = __builtin_amdgcn_wmma_f32_16x16x4_f32(false, Aim[ti], false, Bre[tj], (short)0, S[ti][tj], false, false);
            }
    }

    // C/D 16x16 f32 layout: VGPR r, lane l -> row m = r + 8*(l>>4), col n = l&15
    const int mrow = (lane >> 4) * 8;
#pragma unroll
    for (int ti = 0; ti < 2; ++ti)
#pragma unroll
        for (int tj = 0; tj < 2; ++tj)
#pragma unroll
            for (int r = 0; r < 8; ++r) {
                const int b = b0 + ti * 16 + mrow + r;
                const int f = f0 + tj * 16 + ml;
                float2 o = make_float2(P[ti][tj][r] - Q[ti][tj][r],
                                       R[ti][tj][r] + S[ti][tj][r]);
                *(float2*)(OUT + ((size_t)b * NFEAT + f) * 2) = o;
            }
}

extern "C" void kernel_launch(void* const* d_in, const int* in_sizes, int n_in,
                              void* d_out, int out_size, void* d_ws, size_t ws_size,
                              hipStream_t stream) {
    const float* X  = (const float*)d_in[0];  // cmplx  (2048,512,2)
    const float* A0 = (const float*)d_in[1];  // (64,256)
    const float* A1 = (const float*)d_in[2];  // (64,256)
    const float* B0 = (const float*)d_in[3];  // (64,255)
    const float* B1 = (const float*)d_in[4];  // (64,255)
    const float* om = (const float*)d_in[5];  // (512,)

    float* Ure = (float*)d_ws;                         // 1 MB
    float* Uim = Ure + (size_t)NFEAT * NFEAT;          // 1 MB

    build_u_kernel<<<NFEAT, NPAIR, 0, stream>>>(A0, A1, B0, B1, om, Ure, Uim);
    // 1024 strips of 32x32, 8 waves (256 threads) per block -> 128 blocks
    gemm_kernel<<<128, 256, 0, stream>>>(X, Ure, Uim, (float*)d_out);
}